// MQCNN_Conv2d_4870492914094
// MI455X (gfx1250) — compile-verified
//
#include <hip/hip_runtime.h>
#include <math.h>

typedef __attribute__((ext_vector_type(2))) float v2f;
typedef __attribute__((ext_vector_type(8))) float v8f;

// ---------------- workspace layout (in floats) ----------------
// Bmat : 10*8*256      = 20480    (U matrices, pair-interleaved for b64 loads)
// F    : 10*16*32*1024 = 5242880  (features [o][b][ch][h][w])
// aBuf : 10*16*32      = 5120     (attention gate)
// mean : 320, rstd : 320
#define WS_BMAT 0
#define WS_F    20480
#define WS_A    (20480 + 5242880)
#define WS_MEAN (WS_A + 5120)
#define WS_RSTD (WS_MEAN + 320)

// ---------- compile-time-unrolled 4-qubit gate helpers ----------
// Qubit q <-> bit (3-q) of the flattened state index: q0->8, q1->4, q2->2, q3->1.
template<int MASK>
__device__ inline void ry_g(float st[16], float th) {
  float s, c;
  sincosf(th * 0.5f, &s, &c);
#pragma unroll
  for (int i = 0; i < 16; ++i) {
    if (!(i & MASK)) {
      float lo = st[i], hv = st[i | MASK];
      st[i]        = c * lo - s * hv;
      st[i | MASK] = s * lo + c * hv;
    }
  }
}
template<int CM, int TM>
__device__ inline void cnot_g(float st[16]) {
#pragma unroll
  for (int i = 0; i < 16; ++i) {
    if ((i & CM) && !(i & TM)) {
      float t = st[i]; st[i] = st[i | TM]; st[i | TM] = t;
    }
  }
}
__device__ inline void ry_layer4(float st[16], const float* w) {
  ry_g<8>(st, w[0]); ry_g<4>(st, w[1]); ry_g<2>(st, w[2]); ry_g<1>(st, w[3]);
}
__device__ inline void ring4(float st[16]) {
  cnot_g<8,4>(st); cnot_g<4,2>(st); cnot_g<2,1>(st); cnot_g<1,8>(st);
}

// ---------------- kernel 1: build the 80 circuit unitaries ----------------
// grid 80 (v*16 + j*8 + c), block 16: thread l computes column l of U (= circuit
// applied to basis state e_l), i.e. U[n][l] for n=0..15.
// Storage is packed for the WMMA *A* operand of the main kernel, where lane = n
// and the K (=input state k) pair (2kc2, 2kc2+1) sits in consecutive floats:
//   off(k, n) = (k>>2)*64 + ((k>>1)&1)*32 + n*2 + (k&1)
// so lane n loads a v2f {U[n][k0], U[n][k0+1]} with a single b64.
__global__ void build_u(const float* __restrict__ wq0, const float* __restrict__ wq1,
                        const float* __restrict__ wq2, const float* __restrict__ wq3,
                        const float* __restrict__ wq4, float* __restrict__ Bmat) {
  int blk = blockIdx.x;
  int v = blk >> 4, j = (blk >> 3) & 1, c = blk & 7;
  const float* wp; int K;
  switch (v) {
    case 0: wp = wq0; K = 7;  break;
    case 1: wp = wq1; K = 7;  break;
    case 2: wp = wq2; K = 8;  break;
    case 3: wp = wq3; K = 16; break;
    default: wp = wq4; K = 12; break;
  }
  wp += (j * 8 + c) * K;

  int l = threadIdx.x;     // l = input basis state index (GEMM K index)
  float st[16];
#pragma unroll
  for (int i = 0; i < 16; ++i) st[i] = (i == l) ? 1.0f : 0.0f;

  switch (v) {
    case 0:
      ry_layer4(st, wp); cnot_g<8,4>(st); cnot_g<2,1>(st);
      ry_g<8>(st, wp[4]); ry_g<4>(st, wp[5]); ry_g<2>(st, wp[6]); break;
    case 1:
      ry_layer4(st, wp); cnot_g<8,4>(st); cnot_g<4,2>(st); cnot_g<2,1>(st);
      ry_g<8>(st, wp[4]); ry_g<4>(st, wp[5]); ry_g<2>(st, wp[6]); break;
    case 2:
      ry_layer4(st, wp); ring4(st); ry_layer4(st, wp + 4); break;
    case 3:
      for (int L = 0; L < 4; ++L) { ry_layer4(st, wp + 4 * L); ring4(st); } break;
    default:
      for (int L = 0; L < 3; ++L) { ry_layer4(st, wp + 4 * L); ring4(st); } break;
  }

  int o = v + 5 * j;                       // output feature plane index
  float* dst = Bmat + (o * 8 + c) * 256 + (l >> 2) * 64 + ((l >> 1) & 1) * 32 + (l & 1);
#pragma unroll
  for (int n = 0; n < 16; ++n) dst[n * 2] = st[n];   // element (k=l, n)
}

// ---------------- kernel 2: batched 16x16x16 fp32 WMMA circuit evaluation ----------------
// One wave per tile: (channel c, batch b, row h, 16-wide w tile).
// Roles: D[n][m] = sum_k U[n][k] * init[k][m]
//   A (lane = n, K across VGPRs; documented ISA f32 16x4 A layout) = U rows
//   B (lane = m per half; VGPR0 K=4kc+2hi, VGPR1 K+1)              = init states
// D layout: VGPR r holds state n = r + 8*hi for pixel m = lane&15, so the
// signed probability sums (Pauli-Z expectations) are lane-local with
// compile-time signs; one xor-16 shuffle merges the two state halves.
__global__ __launch_bounds__(32) void qsim_wmma(const float* __restrict__ x,
                                                const float* __restrict__ Bmat,
                                                float* __restrict__ F) {
  int t    = blockIdx.x;
  int c    = t >> 10;
  int rem  = t & 1023;
  int b    = rem >> 6;
  int rem2 = rem & 63;
  int hrow = rem2 >> 1;
  int w0   = (rem2 & 1) << 4;

  int lane = threadIdx.x;
  int hi   = lane >> 4;       // half-wave: K split for A/B operands, state split for D
  int nn   = lane & 15;       // A: state n; B/D: pixel m

  // ---- B operand: per-pixel init state = outer product of (cos, sin) halves ----
  int wcol = w0 + nn;
  const float* xb = x + ((b * 8 + c) * 33 + hrow) * 33 + wcol;
  float a0 = xb[0], a1 = xb[1], a2 = xb[33], a3 = xb[34];
  float c0, s0, c1, s1, c2, s2, c3, s3;
  sincosf(a0 * 0.5f, &s0, &c0); sincosf(a1 * 0.5f, &s1, &c1);
  sincosf(a2 * 0.5f, &s2, &c2); sincosf(a3 * 0.5f, &s3, &c3);
  float p01[4] = {c0 * c1, c0 * s1, s0 * c1, s0 * s1};
  float p23[4] = {c2 * c3, c2 * s3, s2 * c3, s2 * s3};
  float ist[16];
#pragma unroll
  for (int k = 0; k < 16; ++k) ist[k] = p01[k >> 2] * p23[k & 3];

  v2f Breg[4];
#pragma unroll
  for (int kc = 0; kc < 4; ++kc) {
    float e0 = hi ? ist[4 * kc + 2] : ist[4 * kc + 0];
    float e1 = hi ? ist[4 * kc + 3] : ist[4 * kc + 1];
    Breg[kc] = (v2f){e0, e1};
  }

  int hw = hrow * 32 + w0;
  for (int o = 0; o < 10; ++o) {
    // A operand: lane nn loads its v2f pair {U[nn][k0], U[nn][k0+1]}, k0 = 4kc+2hi
    const v2f* Ub = ((const v2f*)(Bmat + (o * 8 + c) * 256)) + hi * 16 + nn;
    v8f acc = {0.f, 0.f, 0.f, 0.f, 0.f, 0.f, 0.f, 0.f};
#pragma unroll
    for (int kc = 0; kc < 4; ++kc) {
      v2f Av = Ub[kc * 32];
      acc = __builtin_amdgcn_wmma_f32_16x16x4_f32(
          /*neg_a=*/false, Av, /*neg_b=*/false, Breg[kc],
          /*c_mod=*/(short)0, acc, /*reuse_a=*/false, /*reuse_b=*/false);
    }

    // Lane-local signed probability sums: n = r + 8*hi
    //   zs[q] = sum_n (-1)^{bit_{3-q}(n)} |amp_n|^2
    // q=1..3 signs depend only on r (compile-time); q=0 sign is the hi bit.
    float p0 = 0.f, p1 = 0.f, p2 = 0.f, p3 = 0.f;
#pragma unroll
    for (int r = 0; r < 8; ++r) {
      float pv = acc[r] * acc[r];
      p0 += pv;
      p1 = (r & 4) ? p1 - pv : p1 + pv;
      p2 = (r & 2) ? p2 - pv : p2 + pv;
      p3 = (r & 1) ? p3 - pv : p3 + pv;
    }
    p0 = hi ? -p0 : p0;
    float z0 = p0 + __shfl_xor(p0, 16, 32);
    float z1 = p1 + __shfl_xor(p1, 16, 32);
    float z2 = p2 + __shfl_xor(p2, 16, 32);
    float z3 = p3 + __shfl_xor(p3, 16, 32);

    // Coalesced stores: 16 consecutive w positions per channel plane.
    if (hi == 0) {
      float* Fo = F + ((o * 16 + b) * 32 + c * 4) * 1024 + hw + nn;
      Fo[0]    = z0;
      Fo[1024] = z1;
      Fo[2048] = z2;
      Fo[3072] = z3;
    }
  }
}

// ---------------- kernel 3: quadrant max-pool + attention MLP ----------------
__global__ __launch_bounds__(128) void attention_k(const float* __restrict__ F,
    const float* __restrict__ W1, const float* __restrict__ B1,
    const float* __restrict__ W2, const float* __restrict__ B2,
    float* __restrict__ aBuf) {
  int o = blockIdx.x / 16, b = blockIdx.x % 16;
  int f = threadIdx.x;
  int ch = f >> 2, qh = (f >> 1) & 1, qw = f & 1;
  const float* Fb = F + ((o * 16 + b) * 32 + ch) * 1024 + qh * 512 + qw * 16;
  float mx = -3.4e38f;
  for (int hh = 0; hh < 16; ++hh)
#pragma unroll 4
    for (int ww = 0; ww < 16; ++ww)
      mx = fmaxf(mx, Fb[hh * 32 + ww]);

  __shared__ float pooled[128];
  __shared__ float hbuf[32];
  pooled[f] = mx;
  __syncthreads();
  if (f < 32) {
    float acc = B1[o * 32 + f];
    const float* wr = W1 + (o * 32 + f) * 128;
    for (int k = 0; k < 128; ++k) acc += pooled[k] * wr[k];
    hbuf[f] = fmaxf(acc, 0.f);
  }
  __syncthreads();
  if (f < 32) {
    float acc = B2[o * 32 + f];
    const float* wr = W2 + (o * 32 + f) * 32;
    for (int k = 0; k < 32; ++k) acc += hbuf[k] * wr[k];
    aBuf[(o * 16 + b) * 32 + f] = 1.f / (1.f + expf(-acc));
  }
}

// ---------------- kernel 4: BN statistics over (B,H,W) per (o, ch) ----------------
__global__ __launch_bounds__(256) void bn_stats(const float* __restrict__ F,
    const float* __restrict__ aBuf, float* __restrict__ meanB, float* __restrict__ rstdB) {
  int o = blockIdx.x >> 5, ch = blockIdx.x & 31;
  int tid = threadIdx.x;
  float sum = 0.f, ssq = 0.f;
  for (int b = 0; b < 16; ++b) {
    float av = aBuf[(o * 16 + b) * 32 + ch];
    float av2 = av * av;
    const float* Fb = F + ((o * 16 + b) * 32 + ch) * 1024;
    for (int s = tid; s < 1024; s += 256) {
      float v = Fb[s];
      sum += av * v;
      ssq += av2 * v * v;
    }
  }
  __shared__ float s1[256], s2[256];
  s1[tid] = sum; s2[tid] = ssq;
  __syncthreads();
  for (int off = 128; off > 0; off >>= 1) {
    if (tid < off) { s1[tid] += s1[tid + off]; s2[tid] += s2[tid + off]; }
    __syncthreads();
  }
  if (tid == 0) {
    float mean = s1[0] * (1.f / 16384.f);
    float var  = s2[0] * (1.f / 16384.f) - mean * mean;
    meanB[o * 32 + ch] = mean;
    rstdB[o * 32 + ch] = rsqrtf(var + 1e-5f);
  }
}

// ---------------- kernel 5: normalize + channel sum + transpose writeout ----------------
__global__ __launch_bounds__(256) void finalize_out(const float* __restrict__ F,
    const float* __restrict__ aBuf, const float* __restrict__ meanB,
    const float* __restrict__ rstdB, const float* __restrict__ gamma,
    const float* __restrict__ beta, float* __restrict__ out) {
  int o = blockIdx.x / 16, b = blockIdx.x % 16;
  int tid = threadIdx.x;
  __shared__ float coef[32], t0[32];
  if (tid < 32) {
    float rs = rstdB[o * 32 + tid];
    float mn = meanB[o * 32 + tid];
    float g = gamma[tid], bt = beta[tid];
    float av = aBuf[(o * 16 + b) * 32 + tid];
    coef[tid] = g * rs * av;
    t0[tid]   = bt - g * rs * mn;
  }
  __syncthreads();
  float base0 = 0.f;
#pragma unroll
  for (int ch = 0; ch < 32; ++ch) base0 += t0[ch];
  const float* Fb = F + ((o * 16 + b) * 32) * 1024;
  for (int s = tid; s < 1024; s += 256) {
    float acc = base0;
#pragma unroll
    for (int ch = 0; ch < 32; ++ch) acc += coef[ch] * Fb[ch * 1024 + s];
    out[((b * 10 + o) << 10) + s] = acc;   // out[b][o][h][w]
  }
}

extern "C" void kernel_launch(void* const* d_in, const int* in_sizes, int n_in,
                              void* d_out, int out_size, void* d_ws, size_t ws_size,
                              hipStream_t stream) {
  (void)in_sizes; (void)n_in; (void)out_size; (void)ws_size;
  const float* x    = (const float*)d_in[0];
  const float* wq0  = (const float*)d_in[1];
  const float* wq1  = (const float*)d_in[2];
  const float* wq2  = (const float*)d_in[3];
  const float* wq3  = (const float*)d_in[4];
  const float* wq4  = (const float*)d_in[5];
  const float* w1   = (const float*)d_in[6];
  const float* b1   = (const float*)d_in[7];
  const float* w2   = (const float*)d_in[8];
  const float* b2   = (const float*)d_in[9];
  const float* gam  = (const float*)d_in[10];
  const float* bet  = (const float*)d_in[11];
  float* ws   = (float*)d_ws;
  float* Bmat = ws + WS_BMAT;
  float* F    = ws + WS_F;
  float* aBuf = ws + WS_A;
  float* mB   = ws + WS_MEAN;
  float* rB   = ws + WS_RSTD;
  float* out  = (float*)d_out;

  build_u<<<80, 16, 0, stream>>>(wq0, wq1, wq2, wq3, wq4, Bmat);
  qsim_wmma<<<8192, 32, 0, stream>>>(x, Bmat, F);
  attention_k<<<160, 128, 0, stream>>>(F, w1, b1, w2, b2, aBuf);
  bn_stats<<<320, 256, 0, stream>>>(F, aBuf, mB, rB);
  finalize_out<<<160, 256, 0, stream>>>(F, aBuf, mB, rB, gam, bet, out);
}